// MultiBoxLoss_55370718380433
// MI455X (gfx1250) — compile-verified
//
#include <hip/hip_runtime.h>
#include <cstdint>

// ---------------- problem constants ----------------
#define B_  32
#define A_  32768
#define G_  50
#define C_  22
#define IOU_TH 0.5f

// ---------------- workspace layout (bytes) ----------------
static constexpr size_t OFF_MINING = 0;                                 // B*A f32 (bg_loss, -inf for pos)
static constexpr size_t OFF_LSE    = OFF_MINING + (size_t)B_*A_*4;      // B*A f32
static constexpr size_t OFF_BIOU   = OFF_LSE    + (size_t)B_*A_*4;      // B*A f32
static constexpr size_t OFF_BIDX   = OFF_BIOU   + (size_t)B_*A_*4;      // B*A i32
static constexpr size_t OFF_CE     = OFF_BIDX   + (size_t)B_*A_*4;      // B*A f32
static constexpr size_t OFF_PACK   = OFF_CE     + (size_t)B_*A_*4;      // B*G u64
static constexpr size_t OFF_NPROW  = OFF_PACK   + (size_t)B_*G_*8;      // B i32
static constexpr size_t OFF_NPTOT  = OFF_NPROW  + (size_t)B_*4;         // i32
static constexpr size_t OFF_LOC    = OFF_NPTOT  + 4;                    // f32
static constexpr size_t OFF_CLS    = OFF_LOC    + 4;                    // f32

// ---------------- CDNA5 async global->LDS staging ----------------
#if defined(__has_builtin)
#  if __has_builtin(__builtin_amdgcn_global_load_async_to_lds_b128)
#    define HAVE_ASYNC_LDS 1
#  endif
#  if __has_builtin(__builtin_amdgcn_s_wait_asynccnt)
#    define HAVE_WAIT_ASYNC 1
#  endif
#endif

typedef int v4i __attribute__((vector_size(16)));
typedef __attribute__((address_space(1))) v4i GV4i;   // global (AS1, prints as __device__)
typedef __attribute__((address_space(3))) v4i LV4i;   // LDS   (AS3)

__device__ __forceinline__ void async_copy16(const float* gsrc, float* ldst) {
#if defined(HAVE_ASYNC_LDS)
  __builtin_amdgcn_global_load_async_to_lds_b128((GV4i*)gsrc, (LV4i*)ldst, 0, 0);
#else
  *(float4*)ldst = *(const float4*)gsrc;
#endif
}

__device__ __forceinline__ void async_wait_all() {
#if defined(HAVE_WAIT_ASYNC)
  __builtin_amdgcn_s_wait_asynccnt(0);
#elif defined(HAVE_ASYNC_LDS)
  asm volatile("s_wait_asynccnt 0" ::: "memory");
#endif
}

// ---------------- wave32 helpers ----------------
__device__ __forceinline__ unsigned long long wave_max_u64(unsigned long long v) {
  #pragma unroll
  for (int off = 16; off > 0; off >>= 1) {
    unsigned long long o = __shfl_down(v, (unsigned)off);
    v = (o > v) ? o : v;
  }
  return v;
}
__device__ __forceinline__ float wave_sum_f32(float v) {
  #pragma unroll
  for (int off = 16; off > 0; off >>= 1) v += __shfl_down(v, (unsigned)off);
  return v;
}
__device__ __forceinline__ float smooth_l1(float x) {
  float d = fabsf(x);
  return (d < 1.f) ? 0.5f * d * d : d - 0.5f;
}

// ---------------- kernel A (first in TU so disasm snippet shows async ops):
// IoU match + log-softmax stats, confidence tile staged via async global->LDS
__global__ __launch_bounds__(256) void k_match_softmax(
    const float* __restrict__ conf, const float* __restrict__ gts,
    const int* __restrict__ counts, const float* __restrict__ anchors,
    float* __restrict__ mining, float* __restrict__ lseArr,
    float* __restrict__ biou, int* __restrict__ bidxArr,
    unsigned long long* __restrict__ packedArr) {
  __shared__ alignas(16) float sConf[256 * C_];   // 22 KB staged tile
  __shared__ float sGts[G_ * 5];

  const int tid = threadIdx.x;
  const int blocksPerRow = A_ / 256;
  const int b  = blockIdx.x / blocksPerRow;
  const int a0 = (blockIdx.x % blocksPerRow) * 256;
  const int a  = a0 + tid;

  // stage this block's confidence tile (256 anchors x 22 classes) via async DMA
  const float* gbase = conf + ((size_t)b * A_ + a0) * C_;
  const int nvec = 256 * C_ / 4;   // 1408 x b128
  #pragma unroll
  for (int i = 0; i < 6; i++) {
    int idx = tid + i * 256;
    if (idx < nvec) async_copy16(gbase + idx * 4, sConf + idx * 4);
  }
  if (tid < G_ * 5) sGts[tid] = gts[(size_t)b * G_ * 5 + tid];
  const int cnt = counts[b];
  async_wait_all();
  __syncthreads();

  // log-softmax stats over C=22
  const float* x = sConf + tid * C_;
  float m = x[0];
  #pragma unroll
  for (int j = 1; j < C_; j++) m = fmaxf(m, x[j]);
  float s = 0.f;
  #pragma unroll
  for (int j = 0; j < C_; j++) s += __expf(x[j] - m);
  float lse = m + __logf(s);
  float bg  = lse - x[0];
  const size_t ia = (size_t)b * A_ + a;
  lseArr[ia] = lse;
  mining[ia] = bg > 0.f ? bg : 0.f;   // clamp: keep uint ordering monotonic

  // anchor box / IoU vs up to 50 GTs
  float4 ab = ((const float4*)anchors)[a];
  float ax0 = ab.x, ay0 = ab.y, ax1 = ab.z, ay1 = ab.w;
  float aarea = fmaxf(ax1 - ax0, 0.f) * fmaxf(ay1 - ay0, 0.f);

  float best = -2.f; int bidx = 0;
  for (int g = 0; g < G_; g++) {
    float gx0 = sGts[g*5+0], gy0 = sGts[g*5+1], gx1 = sGts[g*5+2], gy1 = sGts[g*5+3];
    float ix = fmaxf(fminf(ax1, gx1) - fmaxf(ax0, gx0), 0.f);
    float iy = fmaxf(fminf(ay1, gy1) - fmaxf(ay0, gy0), 0.f);
    float inter = ix * iy;
    float garea = fmaxf(gx1 - gx0, 0.f) * fmaxf(gy1 - gy0, 0.f);
    float iou = inter / (aarea + garea - inter + 1e-5f);
    bool valid = g < cnt;
    float v = valid ? iou : -1.0f;
    if (v > best) { best = v; bidx = g; }   // first-max tie-break == jnp.argmax
    // per-GT best anchor: pack (iou_bits, ~anchor) -> max picks lowest index on ties
    unsigned long long pk = valid
        ? ((((unsigned long long)__float_as_uint(iou)) << 32) | (unsigned)(~a))
        : 0ull;
    pk = wave_max_u64(pk);
    if ((tid & 31) == 0 && pk != 0ull)
      atomicMax(&packedArr[b * G_ + g], pk);   // GLOBAL_ATOMIC_MAX_U64
  }
  biou[ia] = best;
  bidxArr[ia] = bidx;
}

// ---------------- kernel 0: zero accumulators ----------------
__global__ void k_init(unsigned long long* packed, int* npRow, int* npTot,
                       float* locSum, float* clsSum) {
  int i = blockIdx.x * blockDim.x + threadIdx.x;
  if (i < B_ * G_) packed[i] = 0ull;
  if (i < B_) npRow[i] = 0;
  if (i == 0) { *npTot = 0; *locSum = 0.f; *clsSum = 0.f; }
}

// ---------------- kernel B: force-match best prior per GT ----------------
__global__ void k_force(const unsigned long long* __restrict__ packed,
                        const int* __restrict__ counts,
                        float* __restrict__ biou, int* __restrict__ bidxArr) {
  int i = blockIdx.x * blockDim.x + threadIdx.x;
  if (i >= B_ * G_) return;
  int b = i / G_, g = i % G_;
  if (g >= counts[b]) return;
  unsigned a = ~(unsigned)(packed[i] & 0xffffffffull);
  if (a >= A_) return;
  size_t ia = (size_t)b * A_ + a;
  bidxArr[ia] = g;
  biou[ia]    = 2.0f;
}

// ---------------- kernel C: labels, CE, loc loss, pos counts ----------------
__global__ __launch_bounds__(256) void k_targets(
    const float* __restrict__ conf, const float* __restrict__ ploc,
    const float* __restrict__ gts, const float* __restrict__ anchors,
    const float* __restrict__ lseArr, const float* __restrict__ biou,
    const int* __restrict__ bidxArr,
    float* __restrict__ mining, float* __restrict__ ceArr,
    int* __restrict__ npRow, int* __restrict__ npTot,
    float* __restrict__ locSum) {
  __shared__ float sGts[G_ * 5];
  const int tid = threadIdx.x;
  const int blocksPerRow = A_ / 256;
  const int b  = blockIdx.x / blocksPerRow;
  const int a  = (blockIdx.x % blocksPerRow) * 256 + tid;
  if (tid < G_ * 5) sGts[tid] = gts[(size_t)b * G_ * 5 + tid];
  __syncthreads();

  const size_t ia = (size_t)b * A_ + a;
  float iou = biou[ia];
  int   gi  = bidxArr[ia];
  int label = (iou < IOU_TH) ? 0 : (int)sGts[gi * 5 + 4];
  float ce  = lseArr[ia] - conf[ia * C_ + label];
  ceArr[ia] = ce;

  bool pos = label > 0;
  float lsum = 0.f;
  if (pos) {
    mining[ia] = __int_as_float(0xFF800000);   // -inf marker (sign bit)
    float gx0 = sGts[gi*5+0], gy0 = sGts[gi*5+1], gx1 = sGts[gi*5+2], gy1 = sGts[gi*5+3];
    float bcx = (gx0 + gx1) * 0.5f, bcy = (gy0 + gy1) * 0.5f;
    float bw = gx1 - gx0, bh = gy1 - gy0;
    float4 ab = ((const float4*)anchors)[a];
    float acx = (ab.x + ab.z) * 0.5f, acy = (ab.y + ab.w) * 0.5f;
    float aw = ab.z - ab.x, ah = ab.w - ab.y;
    float t0 = (bcx - acx) / aw * 10.f;                 // / CENTER_VARIANCE
    float t1 = (bcy - acy) / ah * 10.f;
    float t2 = __logf(fmaxf(bw / aw, 1e-8f)) * 5.f;     // / SIZE_VARIANCE
    float t3 = __logf(fmaxf(bh / ah, 1e-8f)) * 5.f;
    float4 p = ((const float4*)ploc)[ia];
    lsum = smooth_l1(p.x - t0) + smooth_l1(p.y - t1) +
           smooth_l1(p.z - t2) + smooth_l1(p.w - t3);
  }
  lsum = wave_sum_f32(lsum);
  unsigned long long bal = __ballot(pos);
  if ((tid & 31) == 0) {
    if (lsum != 0.f) atomicAdd(locSum, lsum);
    int n = __popcll(bal);
    if (n) { atomicAdd(&npRow[b], n); atomicAdd(npTot, n); }
  }
}

// ---------------- kernel D: per-row radix top-K negative mining + CE sum ----
__global__ __launch_bounds__(1024) void k_mine_sum(
    const float* __restrict__ mining, const float* __restrict__ ceArr,
    const int* __restrict__ npRow, float* __restrict__ clsSum) {
  __shared__ int hist[256];
  __shared__ int scanBuf[1024];
  __shared__ unsigned sPrefix;
  __shared__ int sKrem;

  const int tid = threadIdx.x;
  const int b   = blockIdx.x;
  const int K   = 3 * npRow[b];
  const size_t base = (size_t)b * A_ + (size_t)tid * 32;

  unsigned u[32];
  #pragma unroll
  for (int i = 0; i < 8; i++) {
    float4 v = ((const float4*)(mining + base))[i];
    u[i*4+0] = __float_as_uint(v.x);
    u[i*4+1] = __float_as_uint(v.y);
    u[i*4+2] = __float_as_uint(v.z);
    u[i*4+3] = __float_as_uint(v.w);
  }

  const bool hasNeg = K > 0;
  unsigned T = 0u;
  int remaining = 0;
  if (hasNeg) {
    unsigned prefix = 0; int krem = K;
    #pragma unroll
    for (int pass = 0; pass < 4; pass++) {
      const int shift = 24 - 8 * pass;
      if (tid < 256) hist[tid] = 0;
      __syncthreads();
      #pragma unroll
      for (int i = 0; i < 32; i++) {
        unsigned uu = u[i];
        if (!(uu >> 31)) {                        // candidate (negative anchor)
          bool inNode = (pass == 0) || ((uu >> (shift + 8)) == prefix);
          if (inNode) atomicAdd(&hist[(uu >> shift) & 255], 1);
        }
      }
      __syncthreads();
      if (tid == 0) {
        int cum = 0, sel = 0, newk = krem, d = 255;
        for (; d >= 0; d--) {
          int c = hist[d];
          if (cum + c >= krem) { sel = d; newk = krem - cum; break; }
          cum += c;
        }
        if (d < 0) { sel = 0; newk = krem - (cum - hist[0]); }  // K >= #cands
        sPrefix = (prefix << 8) | (unsigned)sel;
        sKrem   = newk;
      }
      __syncthreads();
      prefix = sPrefix;
      krem   = sKrem;
    }
    T = prefix;
    remaining = krem;
  }

  // stable tie-break: exclusive block scan of tie counts (index order)
  int myTies = 0;
  if (hasNeg) {
    #pragma unroll
    for (int i = 0; i < 32; i++)
      if (!(u[i] >> 31) && u[i] == T) myTies++;
  }
  scanBuf[tid] = myTies;
  __syncthreads();
  for (int off = 1; off < 1024; off <<= 1) {
    int v = (tid >= off) ? scanBuf[tid - off] : 0;
    __syncthreads();
    scanBuf[tid] += v;
    __syncthreads();
  }
  const int excl = scanBuf[tid] - myTies;

  float sum = 0.f; int local = 0;
  const float* ceRow = ceArr + base;
  #pragma unroll
  for (int i = 0; i < 32; i++) {
    unsigned uu = u[i];
    bool inc = false;
    if (uu >> 31) inc = true;                   // positive -> always in mask
    else if (hasNeg) {
      if (uu > T) inc = true;
      else if (uu == T) { if (excl + local < remaining) inc = true; local++; }
    }
    if (inc) sum += ceRow[i];
  }
  sum = wave_sum_f32(sum);
  if ((tid & 31) == 0) atomicAdd(clsSum, sum);
}

// ---------------- kernel E: finalize ----------------
__global__ void k_final(const float* locSum, const float* clsSum,
                        const int* npTot, float* out) {
  if (threadIdx.x == 0 && blockIdx.x == 0) {
    int np = *npTot;
    float denom = 4.f * (np < 1 ? 1.f : (float)np);
    out[0] = *locSum / denom;   // localisation_loss
    out[1] = *clsSum / denom;   // classification_loss
  }
}

// ---------------- launcher ----------------
extern "C" void kernel_launch(void* const* d_in, const int* in_sizes, int n_in,
                              void* d_out, int out_size, void* d_ws, size_t ws_size,
                              hipStream_t stream) {
  const float* conf   = (const float*)d_in[0];
  const float* ploc   = (const float*)d_in[1];
  const float* gts    = (const float*)d_in[2];
  const int*   counts = (const int*)d_in[3];
  const float* anch   = (const float*)d_in[4];

  char* ws = (char*)d_ws;
  float* mining = (float*)(ws + OFF_MINING);
  float* lseA   = (float*)(ws + OFF_LSE);
  float* biou   = (float*)(ws + OFF_BIOU);
  int*   bidx   = (int*)  (ws + OFF_BIDX);
  float* ceA    = (float*)(ws + OFF_CE);
  unsigned long long* packed = (unsigned long long*)(ws + OFF_PACK);
  int*   npRow  = (int*)  (ws + OFF_NPROW);
  int*   npTot  = (int*)  (ws + OFF_NPTOT);
  float* locS   = (float*)(ws + OFF_LOC);
  float* clsS   = (float*)(ws + OFF_CLS);
  float* out    = (float*)d_out;

  k_init<<<(B_ * G_ + 255) / 256, 256, 0, stream>>>(packed, npRow, npTot, locS, clsS);
  k_match_softmax<<<B_ * (A_ / 256), 256, 0, stream>>>(
      conf, gts, counts, anch, mining, lseA, biou, bidx, packed);
  k_force<<<(B_ * G_ + 255) / 256, 256, 0, stream>>>(packed, counts, biou, bidx);
  k_targets<<<B_ * (A_ / 256), 256, 0, stream>>>(
      conf, ploc, gts, anch, lseA, biou, bidx, mining, ceA, npRow, npTot, locS);
  k_mine_sum<<<B_, 1024, 0, stream>>>(mining, ceA, npRow, clsS);
  k_final<<<1, 1, 0, stream>>>(locS, clsS, npTot, out);
}